// PFASModule_19533511262417
// MI455X (gfx1250) — compile-verified
//
#include <hip/hip_runtime.h>
#include <math.h>

typedef __attribute__((ext_vector_type(2))) float v2f;
typedef __attribute__((ext_vector_type(8))) float v8f;

#define NB_PTS   10000
#define KNN      16
#define TILES    (NB_PTS / 16)   // 625
#define C_IN     64
#define BN_EPS   1e-5f

// ---------------------------------------------------------------------------
// Kernel 0: pack points as float4 (x, y, z, |p|^2) + feature-judge MLP
// (Linear->BN->ReLU->Linear->softmax). One thread per point.
// ---------------------------------------------------------------------------
__global__ void prep_kernel(const float* __restrict__ feat,
                            const float* __restrict__ coord,
                            const float* __restrict__ W1,
                            const float* __restrict__ b1,
                            const float* __restrict__ g_,
                            const float* __restrict__ be_,
                            const float* __restrict__ rm_,
                            const float* __restrict__ rv_,
                            const float* __restrict__ W2,
                            const float* __restrict__ b2,
                            float4* __restrict__ pts,
                            float* __restrict__ probs, int n)
{
    int i = blockIdx.x * blockDim.x + threadIdx.x;
    if (i >= n) return;
    float x = coord[3 * i + 0];
    float y = coord[3 * i + 1];
    float z = coord[3 * i + 2];
    float4 p = {x, y, z, x * x + y * y + z * z};
    pts[i] = p;                                   // global_store_b128

    float h[32];
#pragma unroll
    for (int c = 0; c < 32; ++c) h[c] = b1[c];

    const float4* W1v = reinterpret_cast<const float4*>(W1);
    const float* frow = feat + i * C_IN;
    for (int j = 0; j < C_IN; ++j) {
        float f = frow[j];
#pragma unroll
        for (int c4 = 0; c4 < 8; ++c4) {
            float4 w = W1v[j * 8 + c4];
            h[4 * c4 + 0] = fmaf(f, w.x, h[4 * c4 + 0]);
            h[4 * c4 + 1] = fmaf(f, w.y, h[4 * c4 + 1]);
            h[4 * c4 + 2] = fmaf(f, w.z, h[4 * c4 + 2]);
            h[4 * c4 + 3] = fmaf(f, w.w, h[4 * c4 + 3]);
        }
    }

    float l0 = b2[0], l1 = b2[1], l2 = b2[2];
#pragma unroll
    for (int c = 0; c < 32; ++c) {
        float v = (h[c] - rm_[c]) * rsqrtf(rv_[c] + BN_EPS) * g_[c] + be_[c];
        v = fmaxf(v, 0.0f);
        l0 = fmaf(v, W2[c * 3 + 0], l0);
        l1 = fmaf(v, W2[c * 3 + 1], l1);
        l2 = fmaf(v, W2[c * 3 + 2], l2);
    }
    float m  = fmaxf(l0, fmaxf(l1, l2));
    float e0 = __expf(l0 - m), e1 = __expf(l1 - m), e2 = __expf(l2 - m);
    float inv = 1.0f / (e0 + e1 + e2);
    probs[3 * i + 0] = e0 * inv;
    probs[3 * i + 1] = e1 * inv;
    probs[3 * i + 2] = e2 * inv;
}

// ---------------------------------------------------------------------------
// Kernel 1: fused cdist (V_WMMA_F32_16X16X4_F32 gram tiles) + top-16 +
// local PCA (closed-form 3x3 symmetric eigenvalues) + grid blend.
// One wave per 16 rows of one batch.
//   A operand (16x4 f32, loop-invariant):  lanes 0-15 -> (x,y), 16-31 -> (z,0)
//   B operand (4x16 f32, per col tile):    same per-lane pattern on cols
//   C/D layout: lane l holds N = l%16, M = v + 8*(l/16) for VGPR v in 0..7
// Diagonal (self-distance) masking is hoisted behind a uniform ct==rt branch.
// ---------------------------------------------------------------------------
__global__ __launch_bounds__(32) void knn_kernel(
    const float4* __restrict__ pts,
    const float* __restrict__ probs, float* __restrict__ out)
{
    __shared__ __align__(16) float tile[16][16];   // [N][M]
    __shared__ float mrgD[2][16][KNN];
    __shared__ int   mrgI[2][16][KNN];

    const int lane = threadIdx.x;
    const int hh   = lane >> 4;          // lane half: 0 or 1
    const int r    = lane & 15;
    const int b    = blockIdx.x / TILES;
    const int rt   = blockIdx.x % TILES;
    const int rowstart = rt * 16;
    const int base = b * NB_PTS;
    const float FINF = __builtin_inff();

    // loop-invariant A operand (one b128 load)
    float4 pr = pts[base + rowstart + r];
    v2f a;
    a.x = hh ? pr.z : pr.x;
    a.y = hh ? 0.0f : pr.y;

    // loop-invariant row |p|^2 for this lane's 8 M indices (m = v + 8*hh)
    float srow[8];
#pragma unroll
    for (int v = 0; v < 8; ++v) srow[v] = pts[base + rowstart + 8 * hh + v].w;

    // per-lane sorted top-K (ascending); kd[KNN-1] is current worst
    float kd[KNN]; int ki[KNN];
#pragma unroll
    for (int t = 0; t < KNN; ++t) { kd[t] = FINF; ki[t] = 0; }

#pragma unroll 2
    for (int ct = 0; ct < TILES; ++ct) {
        const int c = ct * 16;

        // one b128 load supplies B operand + col |p|^2 for this lane
        float4 pc = pts[base + c + r];
        // prefetch next column tile into L0 (global_prefetch_b8)
        if (ct + 1 < TILES) __builtin_prefetch(&pts[base + c + 16 + r], 0, 3);
        v2f bb;
        bb.x = hh ? pc.z : pc.x;
        bb.y = hh ? 0.0f : pc.y;
        const float sqn = pc.w;

        v8f acc = {};
        acc = __builtin_amdgcn_wmma_f32_16x16x4_f32(
                  false, a, false, bb, (short)0, acc, false, false);

        // d2 = |pi|^2 + |pj|^2 - 2 <pi,pj>, clamped
        float d2v[8];
#pragma unroll
        for (int v = 0; v < 8; ++v)
            d2v[v] = fmaxf(srow[v] + sqn - 2.0f * acc[v], 0.0f);

        // diagonal -> +inf, only in the single tile that contains it (uniform)
        if (ct == rt) {
#pragma unroll
            for (int v = 0; v < 8; ++v)
                if (rowstart + 8 * hh + v == c + r) d2v[v] = FINF;
        }

        float4 s0 = {d2v[0], d2v[1], d2v[2], d2v[3]};
        float4 s1 = {d2v[4], d2v[5], d2v[6], d2v[7]};
        *reinterpret_cast<float4*>(&tile[r][8 * hh + 0]) = s0;
        *reinterpret_cast<float4*>(&tile[r][8 * hh + 4]) = s1;
        __syncthreads();

        // insertion: lane owns row r; lane-half covers tile cols 8*hh..8*hh+7
#pragma unroll
        for (int j = 0; j < 8; ++j) {
            const int n2  = 8 * hh + j;
            float val = tile[n2][r];
            int gcol  = c + n2;
            if (val < kd[KNN - 1]) {
                kd[KNN - 1] = val; ki[KNN - 1] = gcol;
#pragma unroll
                for (int t = KNN - 1; t > 0; --t) {
                    if (kd[t] < kd[t - 1]) {
                        float td = kd[t]; kd[t] = kd[t - 1]; kd[t - 1] = td;
                        int   ti = ki[t]; ki[t] = ki[t - 1]; ki[t - 1] = ti;
                    }
                }
            }
        }
        __syncthreads();
    }

    // merge the two half-lists of each row (lane r and lane r+16)
#pragma unroll
    for (int t = 0; t < KNN; ++t) { mrgD[hh][r][t] = kd[t]; mrgI[hh][r][t] = ki[t]; }
    __syncthreads();

    if (hh == 0) {
        float md[KNN]; int mi[KNN];
        int i0 = 0, i1 = 0;
#pragma unroll
        for (int t = 0; t < KNN; ++t) {
            float d0 = mrgD[0][r][i0], d1 = mrgD[1][r][i1];
            if (d0 <= d1) { md[t] = d0; mi[t] = mrgI[0][r][i0]; ++i0; }
            else          { md[t] = d1; mi[t] = mrgI[1][r][i1]; ++i1; }
        }

        float mean_dist = 0.0f;
#pragma unroll
        for (int t = 0; t < KNN; ++t) mean_dist += sqrtf(md[t]);
        mean_dist *= (1.0f / KNN);
        float density = 1.0f / (mean_dist + 1e-6f);

        // gather neighbors (one b128 each, L2-resident) and build covariance
        float nx[KNN], nyv[KNN], nz[KNN];
        float mx = 0.0f, my = 0.0f, mz = 0.0f;
#pragma unroll
        for (int t = 0; t < KNN; ++t) {
            float4 p = pts[base + mi[t]];
            nx[t] = p.x; nyv[t] = p.y; nz[t] = p.z;
            mx += p.x; my += p.y; mz += p.z;
        }
        mx *= (1.0f / KNN); my *= (1.0f / KNN); mz *= (1.0f / KNN);

        float cxx = 0, cxy = 0, cxz = 0, cyy = 0, cyz = 0, czz = 0;
#pragma unroll
        for (int t = 0; t < KNN; ++t) {
            float dx = nx[t] - mx, dy = nyv[t] - my, dz = nz[t] - mz;
            cxx = fmaf(dx, dx, cxx); cxy = fmaf(dx, dy, cxy);
            cxz = fmaf(dx, dz, cxz); cyy = fmaf(dy, dy, cyy);
            cyz = fmaf(dy, dz, cyz); czz = fmaf(dz, dz, czz);
        }
        const float invKm1 = 1.0f / (KNN - 1);
        cxx *= invKm1; cxy *= invKm1; cxz *= invKm1;
        cyy *= invKm1; cyz *= invKm1; czz *= invKm1;

        // closed-form eigenvalues of a symmetric 3x3 (descending)
        float q  = (cxx + cyy + czz) * (1.0f / 3.0f);
        float p1 = cxy * cxy + cxz * cxz + cyz * cyz;
        float d00 = cxx - q, d11 = cyy - q, d22 = czz - q;
        float p2 = d00 * d00 + d11 * d11 + d22 * d22 + 2.0f * p1;
        float p  = sqrtf(p2 * (1.0f / 6.0f));
        float e0, e1, e2;
        if (p > 1e-12f) {
            float ip  = 1.0f / p;
            float b00 = d00 * ip, b11 = d11 * ip, b22 = d22 * ip;
            float b01 = cxy * ip, b02 = cxz * ip, b12 = cyz * ip;
            float detB = b00 * (b11 * b22 - b12 * b12)
                       - b01 * (b01 * b22 - b12 * b02)
                       + b02 * (b01 * b12 - b11 * b02);
            float rr  = fminf(fmaxf(0.5f * detB, -1.0f), 1.0f);
            float phi = acosf(rr) * (1.0f / 3.0f);
            e0 = q + 2.0f * p * cosf(phi);
            e2 = q + 2.0f * p * cosf(phi + 2.09439510239319549f); // +2*pi/3
            e1 = 3.0f * q - e0 - e2;
        } else {
            e0 = e1 = e2 = q;
        }
        float lin = (e0 - e1 - e2) / (e0 + e1 + e2 + 1e-6f);

        const int gi = base + rowstart + r;
        float pr0 = probs[3 * gi + 0];
        float pr1 = probs[3 * gi + 1];
        float pr2 = probs[3 * gi + 2];
        float tower = (2.0f * density + pr0) * (1.0f / 3.0f);
        float bgp   = (fmaxf(1.0f - lin, 1.0f - density) + pr1) * (1.0f / 3.0f);
        float linep = (2.0f * lin + pr2) * (1.0f / 3.0f);
        // grids: tower*[.1,.1,.1] + bg*[1,1,1] + line*[.3,.3,1.5] + 1e-6
        out[3 * gi + 0] = tower * 0.1f + bgp + linep * 0.3f + 1e-6f;
        out[3 * gi + 1] = tower * 0.1f + bgp + linep * 0.3f + 1e-6f;
        out[3 * gi + 2] = tower * 0.1f + bgp + linep * 1.5f + 1e-6f;
    }
}

// ---------------------------------------------------------------------------
extern "C" void kernel_launch(void* const* d_in, const int* in_sizes, int n_in,
                              void* d_out, int out_size, void* d_ws, size_t ws_size,
                              hipStream_t stream) {
    const float* feat  = (const float*)d_in[0];
    const float* coord = (const float*)d_in[1];
    // d_in[2] = batch indices (unused: equal-size batches)
    const float* W1  = (const float*)d_in[3];
    const float* b1  = (const float*)d_in[4];
    const float* g_  = (const float*)d_in[5];
    const float* be_ = (const float*)d_in[6];
    const float* rm_ = (const float*)d_in[7];
    const float* rv_ = (const float*)d_in[8];
    const float* W2  = (const float*)d_in[9];
    const float* b2  = (const float*)d_in[10];
    float* out = (float*)d_out;

    const int n = in_sizes[2];                 // B*NB = 20000
    float4* pts  = (float4*)d_ws;              // [n] packed (x,y,z,|p|^2)
    float* probs = (float*)(pts + n);          // [n*3]

    prep_kernel<<<(n + 127) / 128, 128, 0, stream>>>(
        feat, coord, W1, b1, g_, be_, rm_, rv_, W2, b2, pts, probs, n);

    const int nblocks = (n / NB_PTS) * TILES;  // B * 625
    knn_kernel<<<nblocks, 32, 0, stream>>>(pts, probs, out);
}